// SpatialCGNLx_18717467476478
// MI455X (gfx1250) — compile-verified
//
#include <hip/hip_runtime.h>
#include <math.h>

#define B_   16
#define CIN  1024
#define CP   512
#define HW   1024                 // 32*32
#define NB   ((size_t)CP * HW)    // per-batch elems of t/p/g/y
#define NZ   ((size_t)CIN * HW)   // per-batch elems of z/x/out
#define NBLK 64                   // reduction blocks per batch

typedef float v2f __attribute__((ext_vector_type(2)));
typedef float v8f __attribute__((ext_vector_type(8)));

__device__ __forceinline__ v8f wmma_f32(v2f a, v2f b, v8f c) {
  // 8 args: (neg_a, A, neg_b, B, c_mod, C, reuse_a, reuse_b)
  return __builtin_amdgcn_wmma_f32_16x16x4_f32(false, a, false, b, (short)0, c,
                                               false, false);
}

// ---------------------------------------------------------------------------
// Fused GEMM: T/P/G[b] (CP x HW) = Wt/Wp/Wg (CP x CIN) * X[b] (CIN x HW)
// Block: 128 threads = 4 waves.  Block tile: 64(M) x 32(N).
// Wave tile: 16(M) x 32(N) for all 3 outputs (6 accumulators).
// ---------------------------------------------------------------------------
__global__ __launch_bounds__(128) void gemm_tpg_kernel(
    const float* __restrict__ X,
    const float* __restrict__ Wt, const float* __restrict__ Wp,
    const float* __restrict__ Wg,
    float* __restrict__ T, float* __restrict__ P, float* __restrict__ G) {
  const int lane = threadIdx.x & 31;
  const int wave = threadIdx.x >> 5;
  const int l15  = lane & 15;
  const int hi   = lane >> 4;   // 0: K=0,1   1: K=2,3
  const int kl   = hi * 2;
  const int b    = blockIdx.z;
  const int n0   = blockIdx.x * 32;
  const int m0   = blockIdx.y * 64 + wave * 16;

  const float* __restrict__ Xb = X + (size_t)b * CIN * HW;
  const size_t arow = (size_t)(m0 + l15) * CIN;

  v8f t0 = {}, t1 = {}, p0 = {}, p1 = {}, g0 = {}, g1 = {};

  for (int k = 0; k < CIN; k += 4) {
    v2f at = *(const v2f*)(Wt + arow + k + kl);
    v2f ap = *(const v2f*)(Wp + arow + k + kl);
    v2f ag = *(const v2f*)(Wg + arow + k + kl);
    const float* xc = Xb + (size_t)(k + kl) * HW;
    v2f b0, b1;
    b0.x = xc[n0 + l15];
    b0.y = xc[HW + n0 + l15];
    b1.x = xc[n0 + 16 + l15];
    b1.y = xc[HW + n0 + 16 + l15];
    t0 = wmma_f32(at, b0, t0);
    t1 = wmma_f32(at, b1, t1);
    p0 = wmma_f32(ap, b0, p0);
    p1 = wmma_f32(ap, b1, p1);
    g0 = wmma_f32(ag, b0, g0);
    g1 = wmma_f32(ag, b1, g1);
  }

  const int rbase = m0 + hi * 8;
  float* __restrict__ Tb = T + (size_t)b * NB;
  float* __restrict__ Pb = P + (size_t)b * NB;
  float* __restrict__ Gb = G + (size_t)b * NB;
#pragma unroll
  for (int v = 0; v < 8; ++v) {
    size_t r = (size_t)(rbase + v) * HW;
    Tb[r + n0 + l15]      = t0[v];
    Tb[r + n0 + 16 + l15] = t1[v];
    Pb[r + n0 + l15]      = p0[v];
    Pb[r + n0 + 16 + l15] = p1[v];
    Gb[r + n0 + l15]      = g0[v];
    Gb[r + n0 + 16 + l15] = g1[v];
  }
}

// ---------------------------------------------------------------------------
// Z[b] (CIN x HW) = Wz (CIN x CP) * Y[b] (CP x HW)
// Block: 128 threads = 4 waves.  Block tile 64(M) x 64(N); wave 16 x 64.
// ---------------------------------------------------------------------------
__global__ __launch_bounds__(128) void gemm_z_kernel(
    const float* __restrict__ Y, const float* __restrict__ Wz,
    float* __restrict__ Z) {
  const int lane = threadIdx.x & 31;
  const int wave = threadIdx.x >> 5;
  const int l15  = lane & 15;
  const int hi   = lane >> 4;
  const int kl   = hi * 2;
  const int b    = blockIdx.z;
  const int n0   = blockIdx.x * 64;
  const int m0   = blockIdx.y * 64 + wave * 16;

  const float* __restrict__ Yb = Y + (size_t)b * NB;
  const size_t arow = (size_t)(m0 + l15) * CP;

  v8f c0 = {}, c1 = {}, c2 = {}, c3 = {};

  for (int k = 0; k < CP; k += 4) {
    v2f a = *(const v2f*)(Wz + arow + k + kl);
    const float* yc = Yb + (size_t)(k + kl) * HW;
    v2f b0, b1, b2, b3;
    b0.x = yc[n0 + l15];       b0.y = yc[HW + n0 + l15];
    b1.x = yc[n0 + 16 + l15];  b1.y = yc[HW + n0 + 16 + l15];
    b2.x = yc[n0 + 32 + l15];  b2.y = yc[HW + n0 + 32 + l15];
    b3.x = yc[n0 + 48 + l15];  b3.y = yc[HW + n0 + 48 + l15];
    c0 = wmma_f32(a, b0, c0);
    c1 = wmma_f32(a, b1, c1);
    c2 = wmma_f32(a, b2, c2);
    c3 = wmma_f32(a, b3, c3);
  }

  const int rbase = m0 + hi * 8;
  float* __restrict__ Zb = Z + (size_t)b * NZ;
#pragma unroll
  for (int v = 0; v < 8; ++v) {
    size_t r = (size_t)(rbase + v) * HW;
    Zb[r + n0 + l15]      = c0[v];
    Zb[r + n0 + 16 + l15] = c1[v];
    Zb[r + n0 + 32 + l15] = c2[v];
    Zb[r + n0 + 48 + l15] = c3[v];
  }
}

// ---------------------------------------------------------------------------
// att partial sums per (batch, block): S0=sum g, S1=sum p*g, S2=sum p^2*g
// ---------------------------------------------------------------------------
__global__ __launch_bounds__(256) void att_partial_kernel(
    const float* __restrict__ P, const float* __restrict__ G,
    float* __restrict__ part) {
  const int b   = blockIdx.y;
  const int tid = threadIdx.x;
  const float* __restrict__ p = P + (size_t)b * NB;
  const float* __restrict__ g = G + (size_t)b * NB;
  float s0 = 0.f, s1 = 0.f, s2 = 0.f;
  const size_t stride = (size_t)gridDim.x * blockDim.x;
  for (size_t i = (size_t)blockIdx.x * blockDim.x + tid; i < NB; i += stride) {
    float pv = p[i], gv = g[i];
    s0 += gv;
    s1 += pv * gv;
    s2 += pv * pv * gv;
  }
  __shared__ float sm0[256], sm1[256], sm2[256];
  sm0[tid] = s0; sm1[tid] = s1; sm2[tid] = s2;
  __syncthreads();
  for (int off = 128; off > 0; off >>= 1) {
    if (tid < off) {
      sm0[tid] += sm0[tid + off];
      sm1[tid] += sm1[tid + off];
      sm2[tid] += sm2[tid + off];
    }
    __syncthreads();
  }
  if (tid == 0) {
    size_t o = ((size_t)b * gridDim.x + blockIdx.x) * 3;
    part[o] = sm0[0]; part[o + 1] = sm1[0]; part[o + 2] = sm2[0];
  }
}

__global__ void att_final_kernel(const float* __restrict__ part,
                                 float* __restrict__ att, int nblk) {
  int idx = threadIdx.x;  // b*3 + k
  if (idx < B_ * 3) {
    int b = idx / 3, k = idx % 3;
    float s = 0.f;
    for (int j = 0; j < nblk; ++j) s += part[((size_t)b * nblk + j) * 3 + k];
    att[idx] = s;
  }
}

// y = c0*S0 + c1*S1*t + c2*S2*t^2   (ck = alpha_k^2)
__global__ __launch_bounds__(256) void y_kernel(
    const float* __restrict__ T, const float* __restrict__ att,
    float* __restrict__ Y, float c0, float c1, float c2) {
  const int b = blockIdx.y;
  const float s0 = att[b * 3 + 0] * c0;
  const float s1 = att[b * 3 + 1] * c1;
  const float s2 = att[b * 3 + 2] * c2;
  const float* __restrict__ t = T + (size_t)b * NB;
  float* __restrict__ y = Y + (size_t)b * NB;
  const size_t stride = (size_t)gridDim.x * blockDim.x;
  for (size_t i = (size_t)blockIdx.x * blockDim.x + threadIdx.x; i < NB;
       i += stride) {
    float tv = t[i];
    y[i] = s0 + tv * (s1 + tv * s2);
  }
}

// GroupNorm stats partials over z: sum, sum of squares
__global__ __launch_bounds__(256) void stats_partial_kernel(
    const float* __restrict__ Z, float* __restrict__ part) {
  const int b   = blockIdx.y;
  const int tid = threadIdx.x;
  const float* __restrict__ z = Z + (size_t)b * NZ;
  float s = 0.f, ss = 0.f;
  const size_t stride = (size_t)gridDim.x * blockDim.x;
  for (size_t i = (size_t)blockIdx.x * blockDim.x + tid; i < NZ; i += stride) {
    float zv = z[i];
    s += zv;
    ss += zv * zv;
  }
  __shared__ float sm0[256], sm1[256];
  sm0[tid] = s; sm1[tid] = ss;
  __syncthreads();
  for (int off = 128; off > 0; off >>= 1) {
    if (tid < off) {
      sm0[tid] += sm0[tid + off];
      sm1[tid] += sm1[tid + off];
    }
    __syncthreads();
  }
  if (tid == 0) {
    size_t o = ((size_t)b * gridDim.x + blockIdx.x) * 2;
    part[o] = sm0[0]; part[o + 1] = sm1[0];
  }
}

__global__ void stats_final_kernel(const float* __restrict__ part,
                                   float* __restrict__ st, int nblk) {
  int idx = threadIdx.x;  // b*2 + k
  if (idx < B_ * 2) {
    int b = idx / 2, k = idx % 2;
    float s = 0.f;
    for (int j = 0; j < nblk; ++j) s += part[((size_t)b * nblk + j) * 2 + k];
    st[idx] = s;
  }
}

// out = (z - mu) * rsqrt(var + eps) * gn_w[c] + gn_b[c] + x
__global__ __launch_bounds__(256) void final_kernel(
    const float* __restrict__ Z, const float* __restrict__ X,
    const float* __restrict__ gnw, const float* __restrict__ gnb,
    const float* __restrict__ st, float* __restrict__ out) {
  const int b = blockIdx.y;
  const float inv_n = 1.0f / (float)NZ;
  const float mu  = st[b * 2 + 0] * inv_n;
  const float var = st[b * 2 + 1] * inv_n - mu * mu;
  const float inv = rsqrtf(var + 1e-5f);
  const float* __restrict__ z = Z + (size_t)b * NZ;
  const float* __restrict__ x = X + (size_t)b * NZ;
  float* __restrict__ o = out + (size_t)b * NZ;
  const size_t stride = (size_t)gridDim.x * blockDim.x;
  for (size_t i = (size_t)blockIdx.x * blockDim.x + threadIdx.x; i < NZ;
       i += stride) {
    int c = (int)(i / HW);
    o[i] = (z[i] - mu) * inv * gnw[c] + gnb[c] + x[i];
  }
}

extern "C" void kernel_launch(void* const* d_in, const int* in_sizes, int n_in,
                              void* d_out, int out_size, void* d_ws,
                              size_t ws_size, hipStream_t stream) {
  (void)in_sizes; (void)n_in; (void)out_size; (void)ws_size;
  const float* x   = (const float*)d_in[0];
  const float* Wt  = (const float*)d_in[1];
  const float* Wp  = (const float*)d_in[2];
  const float* Wg  = (const float*)d_in[3];
  const float* Wz  = (const float*)d_in[4];
  const float* gnw = (const float*)d_in[5];
  const float* gnb = (const float*)d_in[6];
  float* out = (float*)d_out;
  float* ws  = (float*)d_ws;

  // workspace layout (floats)
  float* T     = ws;                              // [B, CP, HW]
  float* P     = ws + NB * B_;                    // [B, CP, HW]
  float* G     = ws + 2 * NB * B_;                // [B, CP, HW]
  float* Y     = P;                               // alias: p dead after att
  float* Z     = ws + 3 * NB * B_;                // [B, CIN, HW]
  float* APART = Z + NZ * B_;                     // [B, NBLK, 3]
  float* ATT   = APART + (size_t)B_ * NBLK * 3;   // [B, 3]
  float* SPART = ATT + B_ * 3;                    // [B, NBLK, 2]
  float* ST    = SPART + (size_t)B_ * NBLK * 2;   // [B, 2]

  // alpha_k^2 = (2*gamma)^k / k! * exp(-2*gamma)
  const double gamma = 1e-4;
  const double beta  = exp(-2.0 * gamma);
  const float c0 = (float)beta;
  const float c1 = (float)(2.0 * gamma * beta);
  const float c2 = (float)((2.0 * gamma) * (2.0 * gamma) / 2.0 * beta);

  gemm_tpg_kernel<<<dim3(HW / 32, CP / 64, B_), 128, 0, stream>>>(
      x, Wt, Wp, Wg, T, P, G);
  att_partial_kernel<<<dim3(NBLK, B_), 256, 0, stream>>>(P, G, APART);
  att_final_kernel<<<1, 64, 0, stream>>>(APART, ATT, NBLK);
  y_kernel<<<dim3(128, B_), 256, 0, stream>>>(T, ATT, Y, c0, c1, c2);
  gemm_z_kernel<<<dim3(HW / 64, CIN / 64, B_), 128, 0, stream>>>(Y, Wz, Z);
  stats_partial_kernel<<<dim3(NBLK, B_), 256, 0, stream>>>(Z, SPART);
  stats_final_kernel<<<1, 32, 0, stream>>>(SPART, ST, NBLK);
  final_kernel<<<dim3(128, B_), 256, 0, stream>>>(Z, x, gnw, gnb, ST, out);
}